// GATLayer_82970178224598
// MI455X (gfx1250) — compile-verified
//
#include <hip/hip_runtime.h>
#include <math.h>

// GAT layer for MI455X (gfx1250), wave32.
// N=1024, IN_DIM=256, OUT_DIM=128, D_ACT=64, NEG_SLOPE=0.01
// Memory-bound: blend pass moves 768 MB -> ~33us @ 23.3 TB/s.

typedef __attribute__((ext_vector_type(2))) float v2f;
typedef __attribute__((ext_vector_type(4))) float f4;
typedef __attribute__((ext_vector_type(8))) float v8f;

#define NEG_SLOPE 0.01f

__device__ __forceinline__ float lrelu(float x) {
    return (x > 0.f) ? x : NEG_SLOPE * x;
}

// ---------------------------------------------------------------------------
// Kernel 0: build zero-padded B panel Upad[256][16]:
//   Upad[k][0] = sum_o W_fc[o,k] * W_attn[o]        (u_src)
//   Upad[k][1] = sum_o W_fc[o,k] * W_attn[128+o]    (u_dst)
//   Upad[k][2..15] = 0
// Padded layout lets every WMMA lane load its B element unconditionally
// (no EXEC-mask branches in the hot loop).
// ---------------------------------------------------------------------------
__global__ __launch_bounds__(256) void gat_fold(const float* __restrict__ W_fc,
                                                const float* __restrict__ W_attn,
                                                float* __restrict__ Upad) {
    const int k = threadIdx.x;  // 0..255
    float us = 0.f, ud = 0.f;
    #pragma unroll 4
    for (int o = 0; o < 128; ++o) {
        const float w = W_fc[o * 256 + k];
        us = fmaf(w, W_attn[o], us);
        ud = fmaf(w, W_attn[128 + o], ud);
    }
    float* row = Upad + k * 16;
    row[0] = us;
    row[1] = ud;
    #pragma unroll
    for (int c = 2; c < 16; ++c) row[c] = 0.f;
}

// ---------------------------------------------------------------------------
// Kernel 1: s = h @ U via V_WMMA_F32_16X16X4_F32.
// One wave per 16-row tile of h (64 tiles = 8 blocks x 8 waves).
// A (16x4 f32): lanes 0-15 hold M=lane, {K0,K1}; lanes 16-31 hold {K2,K3}.
// B (4x16 f32): lane n=lane&15; VGPR0/1 = rows K0/K1 (half 0), K2/K3 (half 1);
//               loaded branchlessly from the padded panel.
// D (16x16 f32): VGPR v = row v + 8*(lane>=16), column = lane&15.
// ---------------------------------------------------------------------------
__global__ __launch_bounds__(256) void gat_scores(const float* __restrict__ h,
                                                  const float* __restrict__ Upad,
                                                  float* __restrict__ s_src,
                                                  float* __restrict__ s_dst) {
    const int lane = threadIdx.x & 31;
    const int wave = threadIdx.x >> 5;
    const int tile = blockIdx.x * 8 + wave;   // 0..63
    const int m0   = tile * 16;
    const int half = lane >> 4;               // 0 or 1
    const int l15  = lane & 15;

    const float* ha = h + (m0 + l15) * 256 + half * 2;       // A: += 4 per step
    const float* bb = Upad + (half * 2) * 16 + l15;          // B: += 64 per step

    v8f acc = {};
    #pragma unroll 4
    for (int k0 = 0; k0 < 256; k0 += 4) {
        v2f a;
        a.x = ha[k0];
        a.y = ha[k0 + 1];
        v2f b;
        b.x = bb[k0 * 16];        // row k0+2*half,   col l15
        b.y = bb[k0 * 16 + 16];   // row k0+2*half+1, col l15
        acc = __builtin_amdgcn_wmma_f32_16x16x4_f32(
            /*neg_a=*/false, a, /*neg_b=*/false, b,
            /*c_mod=*/(short)0, acc, /*reuse_a=*/false, /*reuse_b=*/false);
    }

    if (l15 < 2) {
        float* outp = (l15 == 0) ? s_src : s_dst;
        const int rbase = m0 + 8 * half;
        #pragma unroll
        for (int v = 0; v < 8; ++v) outp[rbase + v] = acc[v];
    }
}

// ---------------------------------------------------------------------------
// Kernel 2: per-row softmax stats over e[i,j] = lrelu(s_dst[i] + s_src[j]).
// One wave (32 lanes) per row; shfl_xor tree reductions (wave32).
// ---------------------------------------------------------------------------
__global__ __launch_bounds__(256) void gat_rowstats(const float* __restrict__ s_src,
                                                    const float* __restrict__ s_dst,
                                                    float* __restrict__ m_out,
                                                    float* __restrict__ invl_out) {
    const int lane = threadIdx.x & 31;
    const int row  = blockIdx.x * 8 + (threadIdx.x >> 5);
    const float sd = s_dst[row];

    float mx = -INFINITY;
    for (int j = lane; j < 1024; j += 32)
        mx = fmaxf(mx, lrelu(sd + s_src[j]));
    #pragma unroll
    for (int off = 16; off; off >>= 1)
        mx = fmaxf(mx, __shfl_xor(mx, off, 32));

    float sum = 0.f;
    for (int j = lane; j < 1024; j += 32)
        sum += __expf(lrelu(sd + s_src[j]) - mx);
    #pragma unroll
    for (int off = 16; off; off >>= 1)
        sum += __shfl_xor(sum, off, 32);

    if (lane == 0) {
        m_out[row]    = mx;
        invl_out[row] = 1.0f / sum;
    }
}

// ---------------------------------------------------------------------------
// Kernel 3: streaming blend. One float4 per thread (64 floats per (i,j) pair
// -> 16 threads per pair). Recompute alpha from the ~20 KB of cached vectors;
// action/policy/out streamed with nontemporal b128 to bypass cache pollution.
// ---------------------------------------------------------------------------
__global__ __launch_bounds__(256) void gat_blend(const f4* __restrict__ action,
                                                 const f4* __restrict__ policy,
                                                 const float* __restrict__ s_src,
                                                 const float* __restrict__ s_dst,
                                                 const float* __restrict__ m,
                                                 const float* __restrict__ invl,
                                                 f4* __restrict__ out) {
    const int g    = blockIdx.x * 256 + threadIdx.x;  // 0 .. 16M-1
    const int pair = g >> 4;                          // (i,j), 16 float4 per pair
    const int i    = pair >> 10;
    const int j    = pair & 1023;

    const float e     = lrelu(s_dst[i] + s_src[j]);
    const float alpha = __expf(e - m[i]) * invl[i];

    const f4 av = __builtin_nontemporal_load(action + g);
    const f4 pv = __builtin_nontemporal_load(policy + g);

    f4 r;
    r.x = fmaf(alpha, av.x - pv.x, pv.x);
    r.y = fmaf(alpha, av.y - pv.y, pv.y);
    r.z = fmaf(alpha, av.z - pv.z, pv.z);
    r.w = fmaf(alpha, av.w - pv.w, pv.w);

    __builtin_nontemporal_store(r, out + g);
}

// ---------------------------------------------------------------------------
// Launch
// ---------------------------------------------------------------------------
extern "C" void kernel_launch(void* const* d_in, const int* in_sizes, int n_in,
                              void* d_out, int out_size, void* d_ws, size_t ws_size,
                              hipStream_t stream) {
    const float* h      = (const float*)d_in[0];  // 1024 x 256
    const float* action = (const float*)d_in[1];  // 1024 x 1024 x 64
    const float* policy = (const float*)d_in[2];  // 1024 x 1024 x 64
    const float* W_fc   = (const float*)d_in[3];  // 128 x 256
    const float* W_attn = (const float*)d_in[4];  // 1 x 256

    float* ws    = (float*)d_ws;
    float* Upad  = ws;            // 4096 floats (256 x 16, zero-padded B panel)
    float* s_src = ws + 4096;     // 1024
    float* s_dst = ws + 5120;     // 1024
    float* mrow  = ws + 6144;     // 1024
    float* invl  = ws + 7168;     // 1024  (total 32 KB of d_ws)

    gat_fold<<<1, 256, 0, stream>>>(W_fc, W_attn, Upad);
    gat_scores<<<8, 256, 0, stream>>>(h, Upad, s_src, s_dst);
    gat_rowstats<<<128, 256, 0, stream>>>(s_src, s_dst, mrow, invl);

    // 1024*1024*64 / 4 floats-per-thread / 256 threads-per-block = 65536 blocks
    gat_blend<<<65536, 256, 0, stream>>>((const f4*)action, (const f4*)policy,
                                         s_src, s_dst, mrow, invl, (f4*)d_out);
}